// GraphTransformerBlock_24644522344847
// MI455X (gfx1250) — compile-verified
//
#include <hip/hip_runtime.h>
#include <math.h>

typedef __attribute__((ext_vector_type(2))) float v2f;
typedef __attribute__((ext_vector_type(8))) float v8f;

#define HID   128
#define HEADS 4
#define CH    32
#define EDIM  32

__device__ __forceinline__ float gelu_tanh(float x) {
  float x3 = x * x * x;
  float t = tanhf(0.7978845608028654f * (x + 0.044715f * x3));
  return 0.5f * x * (1.0f + t);
}

// float atomic max via sign-split int atomics; dest initialized to 0xFF800000 (-inf)
__device__ __forceinline__ void atomicMaxF32(float* addr, float val) {
  if (val >= 0.0f) atomicMax((int*)addr, __float_as_int(val));
  else             atomicMin((unsigned int*)addr, __float_as_uint(val));
}

// ---------------------------------------------------------------------------
// Generic Y[M,128] = act(A[M,128] @ W[128,128] + bias) (+ resid) via f32 WMMA.
// One wave owns one 16x16 output tile; fully unrolled K -> 32 v_wmma per wave.
// ---------------------------------------------------------------------------
__global__ __launch_bounds__(256) void gemm128_wmma(
    const float* __restrict__ A, const float* __restrict__ W,
    const float* __restrict__ bias, const float* __restrict__ resid,
    float* __restrict__ Y, int M, int act) {
  const int lane = threadIdx.x & 31;
  const int wave = threadIdx.x >> 5;
  const int tile = blockIdx.x * 8 + wave;          // 8 waves / block
  const int ntiles = (M >> 4) * 8;                 // (M/16) row tiles x 8 col tiles
  if (tile >= ntiles) return;                      // wave-uniform: EXEC stays all-1s
  const int rt = tile >> 3;
  const int ct = tile & 7;
  const int ml = lane & 15;
  const int kh = (lane >> 4) << 1;                 // K sub-pair: 0 or 2

  const float* arow = A + (size_t)(rt * 16 + ml) * HID;
  const float* wcol = W + ct * 16 + ml;

  v8f c = {};
  #pragma unroll
  for (int k0 = 0; k0 < HID; k0 += 4) {
    v2f a, b;
    a.x = arow[k0 + kh];
    a.y = arow[k0 + kh + 1];
    b.x = wcol[(size_t)(k0 + kh) * HID];
    b.y = wcol[(size_t)(k0 + kh + 1) * HID];
    c = __builtin_amdgcn_wmma_f32_16x16x4_f32(false, a, false, b, (short)0, c,
                                              false, false);
  }

  const int col = ct * 16 + ml;
  const float bv = bias ? bias[col] : 0.0f;
  const int rbase = rt * 16 + ((lane >> 4) << 3);
  #pragma unroll
  for (int r = 0; r < 8; ++r) {
    const int row = rbase + r;
    float y = c[r] + bv;
    if (act) y = gelu_tanh(y);
    if (resid) y += resid[(size_t)row * HID + col];
    Y[(size_t)row * HID + col] = y;
  }
}

// ---------------------------------------------------------------------------
// Zero agg/denom, set amax = -inf bit pattern.
// ---------------------------------------------------------------------------
__global__ void init_kernel(float* __restrict__ agg, float* __restrict__ denom,
                            unsigned int* __restrict__ amax, int n) {
  const int i = blockIdx.x * blockDim.x + threadIdx.x;
  if (i < n * HID) agg[i] = 0.0f;
  if (i < n * HEADS) { denom[i] = 0.0f; amax[i] = 0xFF800000u; }
}

// ---------------------------------------------------------------------------
// Pass 1 over edges: alpha[e,h] = q[dst]·(k[src]+lift)/sqrt(C); atomic segment max.
// One wave per edge, lane = channel. We (32x128) staged in LDS per block.
// ---------------------------------------------------------------------------
__global__ __launch_bounds__(256) void edge_score_kernel(
    const int* __restrict__ ei, const float* __restrict__ edge_attr,
    const float* __restrict__ We, const float* __restrict__ q,
    const float* __restrict__ k, float* __restrict__ alpha,
    float* __restrict__ amax, int E) {
  __shared__ float sWe[EDIM * HID];
  for (int i = threadIdx.x; i < EDIM * HID; i += blockDim.x) sWe[i] = We[i];
  __syncthreads();

  const int lane = threadIdx.x & 31;
  const int e = blockIdx.x * 8 + (threadIdx.x >> 5);
  if (e >= E) return;
  const int s = ei[e];
  const int d = ei[E + e];

  const float ea = edge_attr[(size_t)e * EDIM + lane];
  float lift[HEADS] = {0.f, 0.f, 0.f, 0.f};
  #pragma unroll 8
  for (int j = 0; j < EDIM; ++j) {
    const float eaj = __shfl(ea, j, 32);
    #pragma unroll
    for (int h = 0; h < HEADS; ++h) lift[h] += eaj * sWe[j * HID + h * CH + lane];
  }

  float acc[HEADS];
  #pragma unroll
  for (int h = 0; h < HEADS; ++h)
    acc[h] = q[(size_t)d * HID + h * CH + lane] *
             (k[(size_t)s * HID + h * CH + lane] + lift[h]);

  for (int off = 16; off; off >>= 1) {
    #pragma unroll
    for (int h = 0; h < HEADS; ++h) acc[h] += __shfl_xor(acc[h], off, 32);
  }

  if (lane == 0) {
    const float scale = 0.17677669529663687f;  // 1/sqrt(32)
    #pragma unroll
    for (int h = 0; h < HEADS; ++h) {
      const float av = acc[h] * scale;
      alpha[(size_t)e * HEADS + h] = av;
      atomicMaxF32(&amax[(size_t)d * HEADS + h], av);
    }
  }
}

// ---------------------------------------------------------------------------
// Pass 2 over edges: ex = exp(alpha - amax[dst]); denom += ex;
// agg[dst] += ex * (v[src] + lift). Normalization deferred to combine kernel.
// ---------------------------------------------------------------------------
__global__ __launch_bounds__(256) void edge_agg_kernel(
    const int* __restrict__ ei, const float* __restrict__ edge_attr,
    const float* __restrict__ We, const float* __restrict__ v,
    const float* __restrict__ alpha, const float* __restrict__ amax,
    float* __restrict__ denom, float* __restrict__ agg, int E) {
  __shared__ float sWe[EDIM * HID];
  for (int i = threadIdx.x; i < EDIM * HID; i += blockDim.x) sWe[i] = We[i];
  __syncthreads();

  const int lane = threadIdx.x & 31;
  const int e = blockIdx.x * 8 + (threadIdx.x >> 5);
  if (e >= E) return;
  const int s = ei[e];
  const int d = ei[E + e];

  const float ea = edge_attr[(size_t)e * EDIM + lane];
  float lift[HEADS] = {0.f, 0.f, 0.f, 0.f};
  #pragma unroll 8
  for (int j = 0; j < EDIM; ++j) {
    const float eaj = __shfl(ea, j, 32);
    #pragma unroll
    for (int h = 0; h < HEADS; ++h) lift[h] += eaj * sWe[j * HID + h * CH + lane];
  }

  float ex[HEADS];
  #pragma unroll
  for (int h = 0; h < HEADS; ++h)
    ex[h] = __expf(alpha[(size_t)e * HEADS + h] - amax[(size_t)d * HEADS + h]);

  #pragma unroll
  for (int h = 0; h < HEADS; ++h)
    atomicAdd(&agg[(size_t)d * HID + h * CH + lane],
              ex[h] * (v[(size_t)s * HID + h * CH + lane] + lift[h]));

  if (lane == 0) {
    #pragma unroll
    for (int h = 0; h < HEADS; ++h) atomicAdd(&denom[(size_t)d * HEADS + h], ex[h]);
  }
}

// ---------------------------------------------------------------------------
// out = agg / (denom + 1e-16) + skip
// ---------------------------------------------------------------------------
__global__ void combine_kernel(const float* __restrict__ agg,
                               const float* __restrict__ denom,
                               const float* __restrict__ skip,
                               float* __restrict__ out, int n) {
  const int i = blockIdx.x * blockDim.x + threadIdx.x;
  if (i >= n * HID) return;
  const int node = i >> 7;
  const int h = (i >> 5) & 3;
  out[i] = agg[i] / (denom[node * HEADS + h] + 1e-16f) + skip[i];
}

// ---------------------------------------------------------------------------
extern "C" void kernel_launch(void* const* d_in, const int* in_sizes, int n_in,
                              void* d_out, int out_size, void* d_ws, size_t ws_size,
                              hipStream_t stream) {
  const float* x     = (const float*)d_in[0];
  const int*   ei    = (const int*)d_in[1];
  const float* eattr = (const float*)d_in[2];
  const float* Wq    = (const float*)d_in[3];
  const float* bq    = (const float*)d_in[4];
  const float* Wk    = (const float*)d_in[5];
  const float* bk    = (const float*)d_in[6];
  const float* Wv    = (const float*)d_in[7];
  const float* bv    = (const float*)d_in[8];
  const float* We    = (const float*)d_in[9];
  const float* Wskip = (const float*)d_in[10];
  const float* bskip = (const float*)d_in[11];
  const float* W1    = (const float*)d_in[12];
  const float* b1    = (const float*)d_in[13];
  const float* W2    = (const float*)d_in[14];
  const float* b2    = (const float*)d_in[15];

  const int N = in_sizes[0] / HID;
  const int E = in_sizes[1] / 2;

  float* ws    = (float*)d_ws;
  float* q     = ws;
  float* k     = q     + (size_t)N * HID;
  float* v     = k     + (size_t)N * HID;
  float* skip  = v     + (size_t)N * HID;
  float* alpha = skip  + (size_t)N * HID;
  float* amax  = alpha + (size_t)E * HEADS;
  float* denom = amax  + (size_t)N * HEADS;
  float* agg   = denom + (size_t)N * HEADS;
  float* outb  = agg   + (size_t)N * HID;
  float* h1    = outb  + (size_t)N * HID;
  float* outf  = (float*)d_out;

  const int gemmGrid = ((N >> 4) * 8 + 7) / 8;       // 8 waves (tiles) per block
  const int edgeGrid = (E + 7) / 8;                  // 8 edges per block
  const int elemGrid = (N * HID + 255) / 256;

  // Node projections (WMMA f32): q, k, v, skip
  gemm128_wmma<<<gemmGrid, 256, 0, stream>>>(x, Wq,    bq,    nullptr, q,    N, 0);
  gemm128_wmma<<<gemmGrid, 256, 0, stream>>>(x, Wk,    bk,    nullptr, k,    N, 0);
  gemm128_wmma<<<gemmGrid, 256, 0, stream>>>(x, Wv,    bv,    nullptr, v,    N, 0);
  gemm128_wmma<<<gemmGrid, 256, 0, stream>>>(x, Wskip, bskip, nullptr, skip, N, 0);

  init_kernel<<<elemGrid, 256, 0, stream>>>(agg, denom, (unsigned int*)amax, N);

  edge_score_kernel<<<edgeGrid, 256, 0, stream>>>(ei, eattr, We, q, k, alpha, amax, E);
  edge_agg_kernel  <<<edgeGrid, 256, 0, stream>>>(ei, eattr, We, v, alpha, amax,
                                                  denom, agg, E);

  combine_kernel<<<elemGrid, 256, 0, stream>>>(agg, denom, skip, outb, N);

  // Residual MLP (WMMA f32 + fused tanh-GELU); final kernel fuses the +out residual.
  gemm128_wmma<<<gemmGrid, 256, 0, stream>>>(outb, W1, b1, nullptr, h1,   N, 1);
  gemm128_wmma<<<gemmGrid, 256, 0, stream>>>(h1,   W2, b2, outb,    outf, N, 1);
}